// InputEmbedderMultimer_1305670058167
// MI455X (gfx1250) — compile-verified
//
#include <hip/hip_runtime.h>
#include <hip/hip_bf16.h>

typedef float v2f __attribute__((ext_vector_type(2)));
typedef float v4f __attribute__((ext_vector_type(4)));
typedef float v8f __attribute__((ext_vector_type(8)));

#define N_RES   768
#define N_CLUST 128
#define TF_DIM  22
#define MSA_DIM 49
#define C_Z     128
#define C_M     256
#define K_PAD   52      // 13 wmma steps * K=4

// ---------------------------------------------------------------------------
// Small GEMMs: out[r,c] = sum_k tf[r,k] * W[k,c] + b[c]   (K = TF_DIM = 22)
// ---------------------------------------------------------------------------
__global__ void small_gemm_kernel(const float* __restrict__ tf,
                                  const float* __restrict__ W,
                                  const float* __restrict__ b,
                                  float* __restrict__ out,
                                  int C, int total) {
  int g = blockIdx.x * blockDim.x + threadIdx.x;
  if (g >= total) return;
  int r = g / C;
  int c = g - r * C;
  float acc = b[c];
  const float* trow = tf + r * TF_DIM;
#pragma unroll
  for (int k = 0; k < TF_DIM; ++k)
    acc = fmaf(trow[k], W[k * C + c], acc);
  out[g] = acc;
}

// ---------------------------------------------------------------------------
// msa_emb = msa_feat(98304x49) @ w_msa_m(49x256) + b_msa_m + tf_m[row%768]
// One wave per 16-row band; 16 N-tiles of 16x16 via V_WMMA_F32_16X16X4_F32.
// ---------------------------------------------------------------------------
__global__ void __launch_bounds__(256)
msa_emb_kernel(const float* __restrict__ msa,
               const float* __restrict__ Wm,    // 49 x 256
               const float* __restrict__ bm,    // 256
               const float* __restrict__ tfm,   // 768 x 256
               float* __restrict__ out) {       // 98304 x 256
  __shared__ float Wlds[K_PAD * C_M];           // 52 KB, rows 49..51 zeroed

  int tid = threadIdx.x;
  for (int idx = tid; idx < K_PAD * C_M; idx += 256)
    Wlds[idx] = (idx < MSA_DIM * C_M) ? Wm[idx] : 0.0f;
  __syncthreads();

  int lane  = tid & 31;
  int wave  = tid >> 5;
  int gwave = blockIdx.x * 8 + wave;
  int m_base = gwave * 16;
  bool hi   = lane >= 16;
  int mlow  = lane & 15;

  // ---- load A fragments once (16x4 f32 tile per step) -------------------
  // lane L holds row M = L%16; VGPR0/1 hold K = {0,1} (lo lanes) / {2,3} (hi)
  v2f afrag[13];
  const float* arow = msa + (size_t)(m_base + mlow) * MSA_DIM;
#pragma unroll
  for (int s = 0; s < 13; ++s) {
    int k0 = 4 * s + (hi ? 2 : 0);
    v2f a;
    a.x = (k0     < MSA_DIM) ? arow[k0]     : 0.0f;
    a.y = (k0 + 1 < MSA_DIM) ? arow[k0 + 1] : 0.0f;
    afrag[s] = a;
  }

  int ncol = lane & 15;
  for (int nt = 0; nt < 16; ++nt) {
    int n_base = nt * 16;
    v8f acc = {0.f, 0.f, 0.f, 0.f, 0.f, 0.f, 0.f, 0.f};
#pragma unroll
    for (int s = 0; s < 13; ++s) {
      int k0 = 4 * s + (hi ? 2 : 0);
      v2f b;
      b.x = Wlds[k0 * C_M + n_base + ncol];          // row K = k0
      b.y = Wlds[(k0 + 1) * C_M + n_base + ncol];    // row K = k0+1
      acc = __builtin_amdgcn_wmma_f32_16x16x4_f32(
          /*neg_a=*/false, afrag[s], /*neg_b=*/false, b,
          /*c_mod=*/(short)0, acc, /*reuse_a=*/false, /*reuse_b=*/false);
    }
    // Epilogue: D vgpr v holds M = v (lanes 0-15) / v+8 (lanes 16-31), N = lane%16
    int col = n_base + ncol;
    float bias = bm[col];
#pragma unroll
    for (int v = 0; v < 8; ++v) {
      int row = m_base + v + (hi ? 8 : 0);
      int res = row % N_RES;
      out[(size_t)row * C_M + col] = acc[v] + bias + tfm[res * C_M + col];
    }
  }
}

// ---------------------------------------------------------------------------
// pair_emb[i,j,:] = tf_emb_i[i] + tf_emb_j[j] + W_pos[fo] + es*w_ent
//                 + W_chain[fc] + b_relpos        (128 channels, float4)
// Block = 256 threads: one i, 8 consecutive j; 32 lanes x float4 = 128 ch.
// ---------------------------------------------------------------------------
__global__ void __launch_bounds__(256)
pair_emb_kernel(const int* __restrict__ ri,   const int* __restrict__ asym,
                const int* __restrict__ ent,  const int* __restrict__ sym,
                const float* __restrict__ tf_i, const float* __restrict__ tf_j,
                const float* __restrict__ w_relpos,  // 73 x 128
                const float* __restrict__ b_relpos,  // 128
                float* __restrict__ out) {           // 768 x 768 x 128
  int tid  = threadIdx.x;
  int lane = tid & 31;
  int p    = tid >> 5;
  int j    = blockIdx.x * 8 + p;
  int i    = blockIdx.y;

  int off     = ri[i] - ri[j];
  int clipped = min(max(off + 32, 0), 64);
  int fo      = (asym[i] == asym[j]) ? clipped : 65;
  bool esame  = (ent[i] == ent[j]);
  int rs      = sym[i] - sym[j];
  int cchain  = min(max(rs + 2, 0), 4);
  int fc      = esame ? cchain : 5;
  float es    = esame ? 1.0f : 0.0f;

  const v4f* vi  = (const v4f*)(tf_i + (size_t)i * C_Z);
  const v4f* vj  = (const v4f*)(tf_j + (size_t)j * C_Z);
  const v4f* vp  = (const v4f*)(w_relpos + (size_t)fo * C_Z);        // W_pos[fo]
  const v4f* ve  = (const v4f*)(w_relpos + (size_t)66 * C_Z);        // w_ent
  const v4f* vc  = (const v4f*)(w_relpos + (size_t)(67 + fc) * C_Z); // W_chain[fc]
  const v4f* vb  = (const v4f*)b_relpos;

  v4f r = vi[lane] + vj[lane] + vp[lane] + es * ve[lane] + vc[lane] + vb[lane];
  ((v4f*)(out + ((size_t)i * N_RES + j) * C_Z))[lane] = r;
}

// ---------------------------------------------------------------------------
extern "C" void kernel_launch(void* const* d_in, const int* in_sizes, int n_in,
                              void* d_out, int out_size, void* d_ws, size_t ws_size,
                              hipStream_t stream) {
  const int*   residue_index = (const int*)  d_in[0];
  const int*   asym_id       = (const int*)  d_in[1];
  const int*   entity_id     = (const int*)  d_in[2];
  const int*   sym_id        = (const int*)  d_in[3];
  const float* target_feat   = (const float*)d_in[4];
  const float* msa_feat      = (const float*)d_in[5];
  const float* w_tf_z_i      = (const float*)d_in[6];
  const float* b_tf_z_i      = (const float*)d_in[7];
  const float* w_tf_z_j      = (const float*)d_in[8];
  const float* b_tf_z_j      = (const float*)d_in[9];
  const float* w_tf_m        = (const float*)d_in[10];
  const float* b_tf_m        = (const float*)d_in[11];
  const float* w_msa_m       = (const float*)d_in[12];
  const float* b_msa_m       = (const float*)d_in[13];
  const float* w_relpos      = (const float*)d_in[14];
  const float* b_relpos      = (const float*)d_in[15];

  float* ws       = (float*)d_ws;
  float* tf_emb_i = ws;                         // 768*128
  float* tf_emb_j = ws + 98304;                 // 768*128
  float* tf_m     = ws + 196608;                // 768*256

  float* out_msa  = (float*)d_out;                     // 128*768*256
  float* out_pair = out_msa + (size_t)25165824;        // 768*768*128

  // 1) small GEMMs into workspace
  small_gemm_kernel<<<384, 256, 0, stream>>>(target_feat, w_tf_z_i, b_tf_z_i,
                                             tf_emb_i, C_Z, N_RES * C_Z);
  small_gemm_kernel<<<384, 256, 0, stream>>>(target_feat, w_tf_z_j, b_tf_z_j,
                                             tf_emb_j, C_Z, N_RES * C_Z);
  small_gemm_kernel<<<768, 256, 0, stream>>>(target_feat, w_tf_m, b_tf_m,
                                             tf_m, C_M, N_RES * C_M);

  // 2) msa_emb via WMMA (98304/16 = 6144 waves, 8 waves/block -> 768 blocks)
  msa_emb_kernel<<<768, 256, 0, stream>>>(msa_feat, w_msa_m, b_msa_m, tf_m,
                                          out_msa);

  // 3) pair_emb (store-bandwidth bound, float4 stores)
  pair_emb_kernel<<<dim3(96, 768), 256, 0, stream>>>(
      residue_index, asym_id, entity_id, sym_id,
      tf_emb_i, tf_emb_j, w_relpos, b_relpos, out_pair);
}